// InvariantPointAttentionNoPair_23081154248894
// MI455X (gfx1250) — compile-verified
//
#include <hip/hip_runtime.h>
#include <hip/hip_bf16.h>
#include <math.h>
#include <stdint.h>

// ---------------------------------------------------------------------------
// Invariant Point Attention (no pair bias) for MI455X / gfx1250.
// fp32 WMMA (V_WMMA_F32_16X16X4_F32) for every contraction; Tensor Data Mover
// (tensor_load_to_lds + s_wait_tensorcnt) stages all B/weight panels in LDS,
// double-buffered against the WMMA loop in attn@V.
// B=1, S=512, C=384, H=12, SQK=SV=16, PQK=4, PV=8.
// ---------------------------------------------------------------------------

#define S 512
#define C 384
#define H 12
#define SQK 16
#define SV 16
#define PQK 4
#define PV 8
#define NEGL (-10000.0f)
#define SCALAR_SCALE 0.25f /* 16^-0.5 */

typedef __attribute__((ext_vector_type(2))) float v2f;
typedef __attribute__((ext_vector_type(8))) float v8f;
typedef __attribute__((ext_vector_type(4))) unsigned int u32x4;
typedef __attribute__((ext_vector_type(8))) int i32x8;
typedef __attribute__((ext_vector_type(4))) int i32x4;

// The TDM builtin is device-only: probe it on the DEVICE pass, where a miss is
// a hard error (compile-time proof the async-tensor path is in the binary).
// The host pass just needs the code to parse; it never codegens kernel bodies.
#if defined(__HIP_DEVICE_COMPILE__)
#if !__has_builtin(__builtin_amdgcn_tensor_load_to_lds)
#error "gfx1250 Tensor Data Mover builtin not available on device pass"
#endif
#define HAS_TDM 1
#else
#define HAS_TDM 0
#endif

__device__ __forceinline__ v8f wmma_f32(v2f a, v2f b, v8f c) {
  return __builtin_amdgcn_wmma_f32_16x16x4_f32(false, a, false, b, (short)0, c,
                                               false, false);
}

__device__ __forceinline__ void wait_tensorcnt0() {
#if HAS_TDM
  __builtin_amdgcn_s_wait_tensorcnt(0);
#endif
}

// ---------------------------------------------------------------------------
// Tensor Data Mover: DMA a 2D f32 tile (tile_w x tile_h, row stride
// row_stride elems) from global memory into LDS at lds_byte_off.
// Descriptor layout per CDNA5 ISA 8.3/8.4 (D# group0 + group1, groups 2/3
// disabled / zero). Issued by one wave; tracked with TENSORcnt.
// ---------------------------------------------------------------------------
__device__ __forceinline__ void tdm_load_f32_2d(unsigned int lds_byte_off,
                                                const float* gsrc,
                                                unsigned int tile_w,
                                                unsigned int tile_h,
                                                unsigned long long row_stride,
                                                unsigned long long tensor_w,
                                                unsigned int tensor_h) {
#if HAS_TDM
  const unsigned long long ga = (unsigned long long)(uintptr_t)gsrc;
  u32x4 g0;
  g0[0] = 1u;                                            // count=1 (valid D#)
  g0[1] = lds_byte_off;                                  // LDS dest (bytes)
  g0[2] = (unsigned int)ga;                              // global_addr[31:0]
  g0[3] = (unsigned int)((ga >> 32) & 0x01ffffffu) | (2u << 30); // addr[56:32] | type=2
  i32x8 g1;
  g1[0] = (int)(2u << 16);                               // data_size=4B; no mcast
  g1[1] = (int)((tensor_w & 0xffffu) << 16);             // tensor_dim0[15:0]
  g1[2] = (int)(((tensor_w >> 16) & 0xffffu) |
                ((unsigned int)(tensor_h & 0xffffu) << 16)); // dim0[31:16]|dim1[15:0]
  g1[3] = (int)(((tensor_h >> 16) & 0xffffu) |
                ((tile_w & 0xffffu) << 16));             // dim1[31:16]|tile_dim0
  g1[4] = (int)(tile_h & 0xffffu);                       // tile_dim1; tile_dim2=0
  g1[5] = (int)(row_stride & 0xffffffffu);               // tensor_dim0_stride[31:0]
  g1[6] = (int)((row_stride >> 32) & 0xffffu);           // stride[47:32]; dim1_stride=0
  g1[7] = 0;
  i32x4 gz = {0, 0, 0, 0};
#if __clang_major__ >= 23
  i32x8 gz8 = {};
  __builtin_amdgcn_tensor_load_to_lds(g0, g1, gz, gz, gz8, 0);
#else
  __builtin_amdgcn_tensor_load_to_lds(g0, g1, gz, gz, 0);
#endif
#else
  (void)lds_byte_off; (void)gsrc; (void)tile_w; (void)tile_h;
  (void)row_stride; (void)tensor_w; (void)tensor_h;
#endif
}

// ---------------------------------------------------------------------------
// Weight-stationary GEMM: Cmat[M,N] = A[M,K] @ W[K,N] + bias, optional row
// mask. 256 threads = 8 waves; each wave one 16x16 tile (8 M-tiles/block).
// The 16-wide W panel (K x 16) is TDM-staged into LDS once per block.
// grid = (M/128, N/16); dynamic LDS = K*16*4 bytes.
// ---------------------------------------------------------------------------
__global__ void gemm_bias_wmma_lds(const float* __restrict__ A,
                                   const float* __restrict__ W,
                                   const float* __restrict__ bias,
                                   const unsigned char* __restrict__ rowmask,
                                   float* __restrict__ Cmat, int N, int K) {
  extern __shared__ float Bs[];            // K x 16 panel
  const int n0 = blockIdx.y << 4;
  if (threadIdx.x < 32) {                  // wave 0 drives the TDM
    tdm_load_f32_2d((unsigned int)(uintptr_t)Bs, W + n0, 16, (unsigned)K,
                    (unsigned long long)N, (unsigned long long)N, (unsigned)K);
    wait_tensorcnt0();
  }
  __syncthreads();

  const int wave = threadIdx.x >> 5;
  const int lane = threadIdx.x & 31;
  const int half = lane >> 4;
  const int l = lane & 15;
  const int m0 = ((blockIdx.x << 3) + wave) << 4;

  const float* arow = A + (size_t)(m0 + l) * K;
  v8f acc = {};
  for (int k = 0; k < K; k += 4) {
    v2f a = *(const v2f*)(arow + k + 2 * half);   // contiguous 8B load
    v2f b;
    b.x = Bs[(k + 2 * half) * 16 + l];
    b.y = Bs[(k + 2 * half + 1) * 16 + l];
    acc = wmma_f32(a, b, acc);
  }
  const float bv = bias[n0 + l];
#pragma unroll
  for (int i = 0; i < 8; ++i) {
    const int m = m0 + i + 8 * half;
    float mask = 1.0f;
    if (rowmask) mask = (rowmask[m] != 0) ? 1.0f : 0.0f;
    Cmat[(size_t)m * N + n0 + l] = (acc[i] + bv) * mask;
  }
}

// ---------------------------------------------------------------------------
// Rotate all point projections into the global frame, in place.
// ---------------------------------------------------------------------------
__global__ void rotate_points_kernel(float* __restrict__ qp,
                                     float* __restrict__ kp,
                                     float* __restrict__ vp,
                                     const float* __restrict__ rot,
                                     const float* __restrict__ trans) {
  const int id = blockIdx.x * blockDim.x + threadIdx.x;
  const int NQ = S * H * PQK;            // 24576
  const int NV = S * H * PV;             // 49152
  if (id >= 2 * NQ + NV) return;

  float* ptr;
  int s;
  if (id < NQ) {
    s = id / (H * PQK);
    ptr = qp + (size_t)s * (H * PQK * 3) + (id % (H * PQK)) * 3;
  } else if (id < 2 * NQ) {
    const int r = id - NQ;
    s = r / (H * PQK);
    ptr = kp + (size_t)s * (H * PQK * 3) + (r % (H * PQK)) * 3;
  } else {
    const int r = id - 2 * NQ;
    s = r / (H * PV);
    ptr = vp + (size_t)s * (H * PV * 3) + (r % (H * PV)) * 3;
  }
  const float* R = rot + (size_t)s * 9;
  const float* t = trans + (size_t)s * 3;
  const float x = ptr[0], y = ptr[1], z = ptr[2];
  ptr[0] = R[0] * x + R[1] * y + R[2] * z + t[0];
  ptr[1] = R[3] * x + R[4] * y + R[5] * z + t[1];
  ptr[2] = R[6] * x + R[7] * y + R[8] * z + t[2];
}

// ---------------------------------------------------------------------------
// Fused per-head Q/K (width 32 = 16 scaled scalar + 12 weight-folded point
// coords + 4 zero pad) and per-key bias bj = 0.5 * sum_p softplus(w)_p|kpg|^2.
// (The row-constant -0.5*ai term cancels in softmax and is dropped.)
// ---------------------------------------------------------------------------
__global__ void prep_qckc_kernel(const float* __restrict__ qs,
                                 const float* __restrict__ ks,
                                 const float* __restrict__ qp,
                                 const float* __restrict__ kp,
                                 const float* __restrict__ pw,
                                 float* __restrict__ Qc,
                                 float* __restrict__ Kc,
                                 float* __restrict__ bj) {
  const int id = blockIdx.x * blockDim.x + threadIdx.x;
  if (id >= H * S) return;
  const int h = id / S;
  const int s = id % S;

  float spw[PQK];
#pragma unroll
  for (int p = 0; p < PQK; ++p) {
    const float x = pw[h * PQK + p];
    spw[p] = (x > 20.0f) ? x : log1pf(expf(x));   // softplus
  }

  float* Q = Qc + (size_t)id * 32;
  float* Kk = Kc + (size_t)id * 32;
  const float* qsr = qs + (size_t)s * (H * SQK) + h * SQK;
  const float* ksr = ks + (size_t)s * (H * SQK) + h * SQK;
#pragma unroll
  for (int d = 0; d < 16; ++d) {
    Q[d] = qsr[d] * SCALAR_SCALE;
    Kk[d] = ksr[d];
  }
  const float* qpr = qp + (size_t)s * (H * PQK * 3) + h * (PQK * 3);
  const float* kpr = kp + (size_t)s * (H * PQK * 3) + h * (PQK * 3);
  float acc = 0.0f;
#pragma unroll
  for (int p = 0; p < PQK; ++p) {
#pragma unroll
    for (int d = 0; d < 3; ++d) {
      const float kv = kpr[p * 3 + d];
      Q[16 + p * 3 + d] = spw[p] * qpr[p * 3 + d];
      Kk[16 + p * 3 + d] = kv;
      acc += spw[p] * kv * kv;
    }
  }
#pragma unroll
  for (int k = 28; k < 32; ++k) { Q[k] = 0.0f; Kk[k] = 0.0f; }
  bj[id] = 0.5f * acc;
}

// ---------------------------------------------------------------------------
// Fused V: per head, width 48 = 16 scalar-V + 24 global point-V + 8 zero pad.
// ---------------------------------------------------------------------------
__global__ void prep_vc_kernel(const float* __restrict__ vs,
                               const float* __restrict__ vp,
                               float* __restrict__ Vc) {
  const int id = blockIdx.x * blockDim.x + threadIdx.x;
  if (id >= H * S) return;
  const int h = id / S;
  const int s = id % S;
  float* V = Vc + (size_t)id * 48;
  const float* vsr = vs + (size_t)s * (H * SV) + h * SV;
#pragma unroll
  for (int d = 0; d < 16; ++d) V[d] = vsr[d];
  const float* vpr = vp + (size_t)s * (H * PV * 3) + h * (PV * 3);
#pragma unroll
  for (int r = 0; r < 24; ++r) V[16 + r] = vpr[r];
#pragma unroll
  for (int r = 40; r < 48; ++r) V[r] = 0.0f;
}

// ---------------------------------------------------------------------------
// Logits: per head L[i,j] = Qc_i . Kc_j - bj[j] (key-masked to -1e4).
// grid = (S/16, S/16, H), one wave per tile; K=32 -> 8 WMMA per tile.
// Both operands are contiguous 8B loads; panels are L2-resident.
// ---------------------------------------------------------------------------
__global__ void logits_wmma_kernel(const float* __restrict__ Qc,
                                   const float* __restrict__ Kc,
                                   const float* __restrict__ bj,
                                   const unsigned char* __restrict__ valid,
                                   float* __restrict__ L) {
  const int lane = threadIdx.x & 31;
  const int half = lane >> 4;
  const int l = lane & 15;
  const int m0 = blockIdx.x << 4;
  const int n0 = blockIdx.y << 4;
  const int h = blockIdx.z;

  const float* qrow = Qc + ((size_t)h * S + m0 + l) * 32;
  const float* krow = Kc + ((size_t)h * S + n0 + l) * 32;
  v8f acc = {};
#pragma unroll
  for (int k = 0; k < 32; k += 4) {
    v2f a = *(const v2f*)(qrow + k + 2 * half);
    v2f b = *(const v2f*)(krow + k + 2 * half);  // Kc^T: row of Kc = col of B
    acc = wmma_f32(a, b, acc);
  }
  const int j = n0 + l;
  const float bjj = bj[h * S + j];
  const bool kv = valid[j] != 0;
  float* Lh = L + (size_t)h * S * S;
#pragma unroll
  for (int i = 0; i < 8; ++i) {
    const int m = m0 + i + 8 * half;
    const float v = acc[i] - bjj;
    Lh[(size_t)m * S + j] = kv ? v : NEGL;
  }
}

// ---------------------------------------------------------------------------
// Row softmax over 512 keys, in place; multiply by query mask afterwards.
// One block (256 threads = 8 wave32) per (h, i) row.
// ---------------------------------------------------------------------------
__global__ void softmax_kernel(float* __restrict__ L,
                               const unsigned char* __restrict__ valid) {
  const int row = blockIdx.x;          // h*S + i
  const int i = row % S;
  float* Lr = L + (size_t)row * S;
  const int tid = threadIdx.x;
  __shared__ float sred[8];

  float m = -3.0e38f;
  for (int j = tid; j < S; j += 256) m = fmaxf(m, Lr[j]);
  for (int o = 16; o > 0; o >>= 1) m = fmaxf(m, __shfl_xor(m, o));
  if ((tid & 31) == 0) sred[tid >> 5] = m;
  __syncthreads();
  float bm = sred[0];
#pragma unroll
  for (int w = 1; w < 8; ++w) bm = fmaxf(bm, sred[w]);
  __syncthreads();

  float sum = 0.0f;
  for (int j = tid; j < S; j += 256) {
    const float p = expf(Lr[j] - bm);
    Lr[j] = p;
    sum += p;
  }
  for (int o = 16; o > 0; o >>= 1) sum += __shfl_xor(sum, o);
  if ((tid & 31) == 0) sred[tid >> 5] = sum;
  __syncthreads();
  float bs = 0.0f;
#pragma unroll
  for (int w = 0; w < 8; ++w) bs += sred[w];

  const float qmask = (valid[i] != 0) ? 1.0f : 0.0f;
  const float inv = qmask / bs;
  for (int j = tid; j < S; j += 256) Lr[j] *= inv;
}

// ---------------------------------------------------------------------------
// Oc[h] (512x48) = attn[h] (512x512) @ Vc[h] (512x48).
// 8 waves / block; per-head V panel TDM-staged into LDS as two 256x48 chunks
// in a 2 x 48 KB double buffer: the DMA for chunk 1 is issued BEFORE the
// chunk-0 WMMA loop and only waited on afterwards (TENSORcnt pipelining).
// Each wave computes a 16x48 strip: A-fragment reused over 3 WMMAs.
// grid = (S/128, H); dynamic LDS = 2*256*48*4 bytes.
// ---------------------------------------------------------------------------
__device__ __forceinline__ void attnv_chunk(const float* __restrict__ Bsc,
                                            const float* __restrict__ arow,
                                            int kc, int half, int l,
                                            v8f& acc0, v8f& acc1, v8f& acc2) {
  for (int k = 0; k < 256; k += 4) {
    v2f a = *(const v2f*)(arow + kc + k + 2 * half);
    const float* brow0 = Bsc + (k + 2 * half) * 48;
    const float* brow1 = brow0 + 48;
    v2f b0, b1, b2;
    b0.x = brow0[l];      b0.y = brow1[l];
    b1.x = brow0[16 + l]; b1.y = brow1[16 + l];
    b2.x = brow0[32 + l]; b2.y = brow1[32 + l];
    acc0 = wmma_f32(a, b0, acc0);
    acc1 = wmma_f32(a, b1, acc1);
    acc2 = wmma_f32(a, b2, acc2);
  }
}

__global__ void attnv_wmma_lds(const float* __restrict__ attn,
                               const float* __restrict__ Vc,
                               float* __restrict__ Oc) {
  extern __shared__ float Bs[];          // 2 x (256 x 48) double buffer
  const int h = blockIdx.y;
  const int wave = threadIdx.x >> 5;
  const int lane = threadIdx.x & 31;
  const int half = lane >> 4;
  const int l = lane & 15;
  const int m0 = ((blockIdx.x << 3) + wave) << 4;
  const bool issuer = (threadIdx.x < 32);

  const float* A = attn + (size_t)h * S * S;
  const float* Bg = Vc + (size_t)h * S * 48;
  const float* arow = A + (size_t)(m0 + l) * S;
  float* Bs1 = Bs + 256 * 48;

  // stage chunk 0 and wait for it
  if (issuer) {
    tdm_load_f32_2d((unsigned int)(uintptr_t)Bs, Bg, 48, 256, 48ull, 48ull, 256);
    wait_tensorcnt0();
  }
  __syncthreads();

  // kick off chunk 1 DMA; it runs while we do the chunk-0 WMMA loop
  if (issuer)
    tdm_load_f32_2d((unsigned int)(uintptr_t)Bs1, Bg + 256 * 48, 48, 256,
                    48ull, 48ull, 256);
  __builtin_prefetch(arow + 256, 0, 1);   // global_prefetch next A strip

  v8f acc0 = {}, acc1 = {}, acc2 = {};
  attnv_chunk(Bs, arow, 0, half, l, acc0, acc1, acc2);

  if (issuer) wait_tensorcnt0();
  __syncthreads();

  attnv_chunk(Bs1, arow, 256, half, l, acc0, acc1, acc2);

  float* Cm = Oc + (size_t)h * S * 48;
#pragma unroll
  for (int i = 0; i < 8; ++i) {
    const int m = m0 + i + 8 * half;
    Cm[(size_t)m * 48 + l]      = acc0[i];
    Cm[(size_t)m * 48 + 16 + l] = acc1[i];
    Cm[(size_t)m * 48 + 32 + l] = acc2[i];
  }
}

// ---------------------------------------------------------------------------
// Assemble combined[s, 576] = [scalar_out(192) | pol(288) | point_norm(96)]
// where pol = R^T (pog - t). One thread per output element.
// ---------------------------------------------------------------------------
__global__ void assemble_kernel(const float* __restrict__ Oc,
                                const float* __restrict__ rot,
                                const float* __restrict__ trans,
                                float* __restrict__ comb) {
  const int id = blockIdx.x * blockDim.x + threadIdx.x;
  if (id >= S * 576) return;
  const int s = id / 576;
  const int c = id % 576;
  const float* R = rot + (size_t)s * 9;
  const float* t = trans + (size_t)s * 3;

  float val;
  if (c < H * SV) {                     // scalar_out
    const int h = c >> 4, d = c & 15;
    val = Oc[((size_t)h * S + s) * 48 + d];
  } else if (c < H * SV + H * PV * 3) { // local-frame points
    const int q = c - H * SV;
    const int h = q / (PV * 3);
    const int r = q % (PV * 3);
    const int p = r / 3, d = r % 3;
    const float* pg = Oc + ((size_t)h * S + s) * 48 + 16 + p * 3;
    const float x0 = pg[0] - t[0], x1 = pg[1] - t[1], x2 = pg[2] - t[2];
    val = R[0 * 3 + d] * x0 + R[1 * 3 + d] * x1 + R[2 * 3 + d] * x2; // R^T x
  } else {                              // point norms
    const int q = c - (H * SV + H * PV * 3);
    const int h = q >> 3, p = q & 7;
    const float* pg = Oc + ((size_t)h * S + s) * 48 + 16 + p * 3;
    const float x0 = pg[0] - t[0], x1 = pg[1] - t[1], x2 = pg[2] - t[2];
    const float l0 = R[0] * x0 + R[3] * x1 + R[6] * x2;
    const float l1 = R[1] * x0 + R[4] * x1 + R[7] * x2;
    const float l2 = R[2] * x0 + R[5] * x1 + R[8] * x2;
    val = sqrtf(l0 * l0 + l1 * l1 + l2 * l2);
  }
  comb[id] = val;
}

// ---------------------------------------------------------------------------
// Host-side launch
// ---------------------------------------------------------------------------
extern "C" void kernel_launch(void* const* d_in, const int* in_sizes, int n_in,
                              void* d_out, int out_size, void* d_ws,
                              size_t ws_size, hipStream_t stream) {
  (void)in_sizes; (void)n_in; (void)out_size; (void)ws_size;

  const float* x      = (const float*)d_in[0];   // (S, C)
  const float* rot    = (const float*)d_in[1];   // (S, 3, 3)
  const float* trans  = (const float*)d_in[2];   // (S, 3)
  const float* Wq_s   = (const float*)d_in[3];
  const float* bq_s   = (const float*)d_in[4];
  const float* Wk_s   = (const float*)d_in[5];
  const float* bk_s   = (const float*)d_in[6];
  const float* Wv_s   = (const float*)d_in[7];
  const float* bv_s   = (const float*)d_in[8];
  const float* Wq_p   = (const float*)d_in[9];
  const float* bq_p   = (const float*)d_in[10];
  const float* Wk_p   = (const float*)d_in[11];
  const float* bk_p   = (const float*)d_in[12];
  const float* Wv_p   = (const float*)d_in[13];
  const float* bv_p   = (const float*)d_in[14];
  const float* pw     = (const float*)d_in[15];  // (H, PQK)
  const float* Wo     = (const float*)d_in[16];  // (576, C)
  const float* bo     = (const float*)d_in[17];
  const unsigned char* valid = (const unsigned char*)d_in[18]; // (S,) bool

  float* out = (float*)d_out;                    // (S, C)

  // workspace carve-up (floats), ~19.2 MB total
  float* ws = (float*)d_ws;
  size_t off = 0;
  float* qs  = ws + off; off += (size_t)S * H * SQK;
  float* ks  = ws + off; off += (size_t)S * H * SQK;
  float* vs  = ws + off; off += (size_t)S * H * SV;
  float* qp  = ws + off; off += (size_t)S * H * PQK * 3;
  float* kp  = ws + off; off += (size_t)S * H * PQK * 3;
  float* vp  = ws + off; off += (size_t)S * H * PV * 3;
  float* Qc  = ws + off; off += (size_t)H * S * 32;
  float* Kc  = ws + off; off += (size_t)H * S * 32;
  float* bj  = ws + off; off += (size_t)H * S;
  float* L   = ws + off; off += (size_t)H * S * S;
  float* Vc  = ws + off; off += (size_t)H * S * 48;
  float* Oc  = ws + off; off += (size_t)H * S * 48;
  float* comb= ws + off; off += (size_t)S * 576;

  // 1) input projections (TDM-staged weight panels, fp32 WMMA)
  gemm_bias_wmma_lds<<<dim3(4, (H * SQK) / 16), 256, C * 16 * 4, stream>>>(x, Wq_s, bq_s, nullptr, qs, H * SQK, C);
  gemm_bias_wmma_lds<<<dim3(4, (H * SQK) / 16), 256, C * 16 * 4, stream>>>(x, Wk_s, bk_s, nullptr, ks, H * SQK, C);
  gemm_bias_wmma_lds<<<dim3(4, (H * SV)  / 16), 256, C * 16 * 4, stream>>>(x, Wv_s, bv_s, nullptr, vs, H * SV, C);
  gemm_bias_wmma_lds<<<dim3(4, (H * PQK * 3) / 16), 256, C * 16 * 4, stream>>>(x, Wq_p, bq_p, nullptr, qp, H * PQK * 3, C);
  gemm_bias_wmma_lds<<<dim3(4, (H * PQK * 3) / 16), 256, C * 16 * 4, stream>>>(x, Wk_p, bk_p, nullptr, kp, H * PQK * 3, C);
  gemm_bias_wmma_lds<<<dim3(4, (H * PV * 3) / 16), 256, C * 16 * 4, stream>>>(x, Wv_p, bv_p, nullptr, vp, H * PV * 3, C);

  // 2) rotate points to global frame (in place)
  {
    const int total = S * H * (PQK + PQK + PV);  // 98304
    rotate_points_kernel<<<(total + 255) / 256, 256, 0, stream>>>(qp, kp, vp, rot, trans);
  }

  // 3) fuse scalar + point channels
  prep_qckc_kernel<<<(H * S + 255) / 256, 256, 0, stream>>>(qs, ks, qp, kp, pw, Qc, Kc, bj);
  prep_vc_kernel<<<(H * S + 255) / 256, 256, 0, stream>>>(vs, vp, Vc);

  // 4) logits (WMMA), 5) softmax, 6) attn @ V (double-buffered TDM + WMMA)
  logits_wmma_kernel<<<dim3(S / 16, S / 16, H), 32, 0, stream>>>(Qc, Kc, bj, valid, L);
  softmax_kernel<<<H * S, 256, 0, stream>>>(L, valid);
  attnv_wmma_lds<<<dim3(S / 128, H), 256, 2 * 256 * 48 * 4, stream>>>(L, Vc, Oc);

  // 7) inverse rotation, norms, concat
  assemble_kernel<<<(S * 576 + 255) / 256, 256, 0, stream>>>(Oc, rot, trans, comb);

  // 8) output projection + row mask
  gemm_bias_wmma_lds<<<dim3(4, C / 16), 256, 576 * 16 * 4, stream>>>(comb, Wo, bo, valid, out, C, 576);
}